// MetaphorModel_30193620091217
// MI455X (gfx1250) — compile-verified
//
#include <hip/hip_runtime.h>

typedef float v2f __attribute__((ext_vector_type(2)));
typedef float v4f __attribute__((ext_vector_type(4)));
typedef float v8f __attribute__((ext_vector_type(8)));

#define BB 256   // batch
#define SS 512   // sequence
#define DD 768   // hidden
#define CC 2     // classes

// ---------------------------------------------------------------------------
// Kernel 1: masked mean pooling  pooled[b,d] = sum_s mask[b,s]*h[b,s,d]/count_b
// One block per sample. 192 threads, each owns one float4 (4 d-columns).
// Streams 1.5 MB per block with non-temporal 128-bit loads (read-once data,
// 403 MB total > 192 MB L2, so keep it out of the cache).
// ---------------------------------------------------------------------------
__global__ __launch_bounds__(192) void pool_kernel(const float* __restrict__ h,
                                                   const unsigned char* __restrict__ mask,
                                                   float* __restrict__ pooled) {
    __shared__ float smask[SS];
    __shared__ float sinv;
    const int b   = blockIdx.x;
    const int tid = threadIdx.x;

    for (int s = tid; s < SS; s += 192)
        smask[s] = mask[b * SS + s] ? 1.0f : 0.0f;
    __syncthreads();
    if (tid == 0) {
        float c = 0.0f;
        for (int s = 0; s < SS; ++s) c += smask[s];
        sinv = 1.0f / c;   // reference guarantees count >= 1
    }
    __syncthreads();

    const v4f* __restrict__ base = (const v4f*)(h + (size_t)b * SS * DD);
    v4f acc = {0.0f, 0.0f, 0.0f, 0.0f};
#pragma unroll 4
    for (int s = 0; s < SS; ++s) {
        v4f x = __builtin_nontemporal_load(base + s * (DD / 4) + tid);
        acc += smask[s] * x;
    }
    const float inv = sinv;
    *(v4f*)(pooled + (size_t)b * DD + tid * 4) = acc * inv;
}

// ---------------------------------------------------------------------------
// Kernel 2: classifier GEMM via V_WMMA_F32_16X16X4_F32 (exact f32 math).
// logits[256,2] = pooled[256,768] @ W^T[768,2] + bias.
// 16 blocks x 1 wave (32 lanes). Each wave computes a 16x16 tile (N padded
// 2 -> 16), sweeping K = 768 in steps of 4 -> 192 v_wmma per wave.
//
// A (16x4 f32), per ISA layout: lane holds A[M=lane%16][K = 2*(lane>>4)+i]
// B ( 4x16 f32), mirrored:      lane holds B[K = 2*(lane>>4)+i][N=lane%16]
// C/D (16x16 f32): VGPR r, lane l -> M = r + 8*(l>>4), N = l%16
// ---------------------------------------------------------------------------
__global__ __launch_bounds__(32) void gemm_kernel(const float* __restrict__ pooled,
                                                  const float* __restrict__ W,    // [C][D]
                                                  const float* __restrict__ bias, // [C]
                                                  float* __restrict__ logits) {   // [B][C]
    const int tile = blockIdx.x;          // 16 tiles of 16 samples
    const int lane = threadIdx.x;         // 0..31
    const int mrow = lane & 15;
    const int pair = lane >> 4;           // 0 or 1

    // Zero padded B columns via select (keeps EXEC all-ones for WMMA).
    const float bsc = (mrow < CC) ? 1.0f : 0.0f;
    const float* __restrict__ arow = pooled + (size_t)(tile * 16 + mrow) * DD;
    const float* __restrict__ brow = W + (size_t)((mrow < CC) ? mrow : 0) * DD;

    v8f acc = {};
    for (int k0 = 0; k0 < DD; k0 += 4) {
        const int k = k0 + 2 * pair;
        v2f a = { arow[k], arow[k + 1] };
        v2f bmat = { brow[k] * bsc, brow[k + 1] * bsc };
        acc = __builtin_amdgcn_wmma_f32_16x16x4_f32(
            /*neg_a=*/false, a, /*neg_b=*/false, bmat,
            /*c_mod=*/(short)0, acc, /*reuse_a=*/false, /*reuse_b=*/false);
    }

    const int n = lane & 15;
    if (n < CC) {
        const float bv = bias[n];
#pragma unroll
        for (int r = 0; r < 8; ++r) {
            const int m = r + 8 * pair;
            logits[(size_t)(tile * 16 + m) * CC + n] = acc[r] + bv;
        }
    }
}

// ---------------------------------------------------------------------------
// Kernel 3: mean cross-entropy over B=256 samples (C=2), LDS tree reduction.
// ---------------------------------------------------------------------------
__global__ __launch_bounds__(256) void loss_kernel(const float* __restrict__ logits,
                                                   const int* __restrict__ labels,
                                                   float* __restrict__ loss) {
    __shared__ float red[256];
    const int t = threadIdx.x;
    const float l0 = logits[t * 2 + 0];
    const float l1 = logits[t * 2 + 1];
    const float m  = fmaxf(l0, l1);
    const float lse = m + logf(expf(l0 - m) + expf(l1 - m));
    const int lab = labels[t];
    red[t] = lse - (lab == 0 ? l0 : l1);
    __syncthreads();
#pragma unroll
    for (int off = 128; off > 0; off >>= 1) {
        if (t < off) red[t] += red[t + off];
        __syncthreads();
    }
    if (t == 0) loss[0] = red[0] * (1.0f / 256.0f);
}

extern "C" void kernel_launch(void* const* d_in, const int* in_sizes, int n_in,
                              void* d_out, int out_size, void* d_ws, size_t ws_size,
                              hipStream_t stream) {
    (void)in_sizes; (void)n_in; (void)out_size; (void)ws_size;

    const float*         h      = (const float*)d_in[0];          // [B,S,D] f32
    const unsigned char* mask   = (const unsigned char*)d_in[1];  // [B,S] bool (1B)
    const int*           labels = (const int*)d_in[2];            // [B] i32
    const float*         W      = (const float*)d_in[3];          // [C,D] f32
    const float*         bias   = (const float*)d_in[4];          // [C] f32

    float* out    = (float*)d_out;   // [0]=loss, [1..512]=logits
    float* logits = out + 1;
    float* pooled = (float*)d_ws;    // [B,D] = 768 KB scratch

    pool_kernel<<<BB, 192, 0, stream>>>(h, mask, pooled);
    gemm_kernel<<<BB / 16, 32, 0, stream>>>(pooled, W, bias, logits);
    loss_kernel<<<1, 256, 0, stream>>>(logits, labels, out);
}